// BertNer_51797305590132
// MI455X (gfx1250) — compile-verified
//
#include <hip/hip_runtime.h>
#include <hip/hip_bf16.h>
#include <math.h>

// Problem dims (fixed by the reference)
#define BATCH 64
#define SEQ   512
#define HID   768
#define LBL   9
#define NPAD  16   // pad L=9 -> 16 for WMMA N dimension

typedef __attribute__((ext_vector_type(2))) float v2f;
typedef __attribute__((ext_vector_type(8))) float v8f;

// ---------------------------------------------------------------------------
// Kernel 0: zero a HID-float "dummy row" in workspace. Invalid compacted rows
// gather from this row so the WMMA loop needs no per-iteration masking.
// ---------------------------------------------------------------------------
__global__ __launch_bounds__(256) void zero_row_kernel(float* __restrict__ zrow)
{
    const int i = blockIdx.x * 256 + threadIdx.x;
    if (i < HID) zrow[i] = 0.0f;
}

// ---------------------------------------------------------------------------
// Kernel 1: per-batch-row stable stream compaction map.
//   src_idx[b*SEQ + d] = s  where token s is the d-th valid token of row b
//   counts[b]          = number of valid tokens in row b
// ---------------------------------------------------------------------------
__global__ __launch_bounds__(SEQ) void compact_map_kernel(
    const int* __restrict__ valid_mask,
    int* __restrict__ src_idx,
    int* __restrict__ counts)
{
    __shared__ int scan[SEQ];
    const int b = blockIdx.x;
    const int s = threadIdx.x;

    const int v = valid_mask[b * SEQ + s];
    scan[s] = v;
    __syncthreads();

    // Hillis–Steele inclusive scan
    for (int off = 1; off < SEQ; off <<= 1) {
        int t = (s >= off) ? scan[s - off] : 0;
        __syncthreads();
        scan[s] += t;
        __syncthreads();
    }

    if (v) src_idx[b * SEQ + (scan[s] - 1)] = s;
    if (s == SEQ - 1) counts[b] = scan[SEQ - 1];
}

// ---------------------------------------------------------------------------
// Kernel 2: gathered GEMM (f32 WMMA 16x16x4) + bias + softmax(L=9).
// One wave handles one 16-row tile of one batch. 8 waves / block.
// ---------------------------------------------------------------------------
__global__ __launch_bounds__(256) void ner_wmma_softmax_kernel(
    const float* __restrict__ X,      // [BATCH, SEQ, HID]
    const float* __restrict__ W,      // [HID, LBL]
    const float* __restrict__ bias,   // [LBL]
    const int*   __restrict__ src_idx,
    const int*   __restrict__ counts,
    const float* __restrict__ zrow,   // HID zeros
    float* __restrict__ out)          // [BATCH, SEQ, LBL]
{
    __shared__ float Wlds[HID * NPAD];       // 48 KB, W padded to 16 cols
    __shared__ float Clds[8 * 16 * NPAD];    // 8 waves * 1 KB C tiles

    const int tid = threadIdx.x;

    // Cooperative load of W into LDS, padded with zeros for n in [9,16)
    for (int i = tid; i < HID * NPAD; i += blockDim.x) {
        const int k = i >> 4;
        const int n = i & 15;
        Wlds[i] = (n < LBL) ? W[k * LBL + n] : 0.0f;
    }
    __syncthreads();

    const int wave = tid >> 5;                 // 0..7
    const int lane = tid & 31;
    const int tile = blockIdx.x * 8 + wave;    // 2048 tiles total
    const int bb   = tile >> 5;                // batch index (tiles/batch = 32)
    const int trow = (tile & 31) << 4;         // tile base row within batch

    const int m    = lane & 15;                // row-of-tile (A) / col (B)
    const int khi  = lane >> 4;                // 0 or 1
    const int koff = khi << 1;                 // 0 or 2

    // Gather: which source token feeds compacted destination row (bb, trow+m).
    // Invalid rows read the shared zero row -> logits = bias -> softmax(bias),
    // exactly matching the reference's zero-filled compaction slots.
    const int  d        = trow + m;
    const int  cnt      = counts[bb];
    const bool validRow = (d < cnt);
    const int  srow     = validRow ? src_idx[bb * SEQ + d] : 0;
    const float* rowp   = validRow ? (X + ((size_t)bb * SEQ + (size_t)srow) * HID)
                                   : zrow;
    const float2* xrow  = (const float2*)rowp;

    v8f c = {0.f, 0.f, 0.f, 0.f, 0.f, 0.f, 0.f, 0.f};

    // K loop: 768 / 4 = 192 WMMA_F32_16X16X4_F32 accumulations.
    // Pure load->wmma body; compiler software-pipelines the b64/ds loads.
    for (int k = 0; k < HID; k += 4) {
        // A fragment (16x4 f32): lane(16*khi+m) holds A[m][k+2*khi +{0,1}]
        float2 av = xrow[(k + koff) >> 1];
        v2f a;
        a.x = av.x;
        a.y = av.y;

        // B fragment (4x16 f32): lane(16*khi+n) holds B[k+2*khi +{0,1}][n]
        v2f bm;
        bm.x = Wlds[(k + koff)     * NPAD + m];
        bm.y = Wlds[(k + koff + 1) * NPAD + m];

        c = __builtin_amdgcn_wmma_f32_16x16x4_f32(
                /*neg_a=*/false, a, /*neg_b=*/false, bm,
                /*c_mod=*/(short)0, c, /*reuse_a=*/false, /*reuse_b=*/false);
    }

    // Spill C tile (+bias) to LDS using documented C layout:
    // VGPR j, lanes 0-15 -> M=j, N=lane; lanes 16-31 -> M=8+j, N=lane-16
    float* cl = Clds + wave * (16 * NPAD);
    const float bn = (m < LBL) ? bias[m] : 0.0f;
    const int mbase = khi * 8;
#pragma unroll
    for (int j = 0; j < 8; ++j) {
        cl[(mbase + j) * NPAD + m] = c[j] + bn;
    }
    __syncthreads();

    // Softmax epilogue: 16 lanes per wave, one compacted row each
    if (lane < 16) {
        const int r = lane;
        float v[LBL];
        float mx = -INFINITY;
#pragma unroll
        for (int n = 0; n < LBL; ++n) {
            v[n] = cl[r * NPAD + n];
            mx = fmaxf(mx, v[n]);
        }
        float sum = 0.0f;
#pragma unroll
        for (int n = 0; n < LBL; ++n) {
            v[n] = expf(v[n] - mx);
            sum += v[n];
        }
        const float inv = 1.0f / sum;
        const size_t obase = ((size_t)bb * SEQ + (size_t)(trow + r)) * LBL;
#pragma unroll
        for (int n = 0; n < LBL; ++n) {
            out[obase + n] = v[n] * inv;
        }
    }
}

// ---------------------------------------------------------------------------
extern "C" void kernel_launch(void* const* d_in, const int* in_sizes, int n_in,
                              void* d_out, int out_size, void* d_ws, size_t ws_size,
                              hipStream_t stream) {
    (void)in_sizes; (void)n_in; (void)out_size; (void)ws_size;

    const float* X    = (const float*)d_in[0];   // [64,512,768] f32
    const int*   mask = (const int*)  d_in[1];   // [64,512] i32
    const float* W    = (const float*)d_in[2];   // [768,9] f32
    const float* bias = (const float*)d_in[3];   // [9] f32
    float*       out  = (float*)d_out;           // [64,512,9] f32

    // Workspace layout: src_idx (64*512 i32), counts (64 i32, padded), zero row
    int*   src_idx = (int*)d_ws;
    int*   counts  = src_idx + BATCH * SEQ;              // at +128 KB
    float* zrow    = (float*)((char*)d_ws + BATCH * SEQ * 4 + 256); // 256B-aligned

    zero_row_kernel<<<(HID + 255) / 256, 256, 0, stream>>>(zrow);
    compact_map_kernel<<<BATCH, SEQ, 0, stream>>>(mask, src_idx, counts);

    // 2048 tiles of 16 rows, 8 waves (tiles) per 256-thread block
    ner_wmma_softmax_kernel<<<(BATCH * (SEQ / 16)) / 8, 256, 0, stream>>>(
        X, W, bias, src_idx, counts, zrow, out);
}